// FeedForwardAttention_51874615001165
// MI455X (gfx1250) — compile-verified
//
#include <hip/hip_runtime.h>
#include <math.h>

typedef float v2f __attribute__((ext_vector_type(2)));
typedef float v8f __attribute__((ext_vector_type(8)));

constexpr int B_ = 8, L_ = 2048, CIN_ = 1024, COUT_ = 1024;
constexpr int MPAD_ = 16;                     // WMMA M tile (rows 8..15 zero)

// workspace layout (float offsets)
constexpr int WS_U      = 0;                        // u[CIN]
constexpr int WS_C0     = CIN_;                     // scalar
constexpr int WS_S      = 1040;                     // scores/probs, B*L
constexpr int WS_VBAR   = WS_S + B_ * L_;           // vbar[MPAD][CIN] (padded)
constexpr int WS_OUTROW = WS_VBAR + MPAD_ * CIN_;   // out_row[B][COUT]

// ---------------------------------------------------------------------------
// k1: u[c] = sum_o fck_w[o]*Wk[o][c];  c0 = fck_w·bk + fck_b;
//     zero padded vbar (16 x CIN).   grid 4 x 256
// ---------------------------------------------------------------------------
__global__ void k1_fold_key_weights(const float* __restrict__ Wk,
                                    const float* __restrict__ bk,
                                    const float* __restrict__ fckw,
                                    const float* __restrict__ fckb,
                                    float* __restrict__ ws) {
  int c = blockIdx.x * 256 + threadIdx.x;          // 0..1023
  float acc = 0.f;
  for (int o = 0; o < COUT_; ++o)
    acc = fmaf(fckw[o], Wk[(size_t)o * CIN_ + c], acc);
  ws[WS_U + c] = acc;

  for (int i = c; i < MPAD_ * CIN_; i += 1024)     // zero all 16 padded rows
    ws[WS_VBAR + i] = 0.f;

  __shared__ float red[256];
  if (blockIdx.x == 0) {
    float pa = 0.f;
    for (int o = threadIdx.x; o < COUT_; o += 256) pa = fmaf(fckw[o], bk[o], pa);
    red[threadIdx.x] = pa;
    __syncthreads();
    for (int stride = 128; stride > 0; stride >>= 1) {
      if ((int)threadIdx.x < stride) red[threadIdx.x] += red[threadIdx.x + stride];
      __syncthreads();
    }
    if (threadIdx.x == 0) ws[WS_C0] = red[0] + fckb[0];
  }
}

// ---------------------------------------------------------------------------
// k2: s[b,l] = key[b,l,:]·u + c0.   One wave32 per row, float4 loads.
// grid B*L/8 x 256
// ---------------------------------------------------------------------------
__global__ void k2_scores(const float* __restrict__ key, float* __restrict__ ws) {
  int lane = threadIdx.x & 31, wv = threadIdx.x >> 5;
  int row = blockIdx.x * 8 + wv;                   // 0..B*L-1
  const float4* krow = (const float4*)(key + (size_t)row * CIN_);
  const float4* u4   = (const float4*)(ws + WS_U);
  float acc = 0.f;
  for (int j = lane; j < CIN_ / 4; j += 32) {
    float4 kv = krow[j];
    float4 uv = u4[j];
    acc = fmaf(kv.x, uv.x, acc);
    acc = fmaf(kv.y, uv.y, acc);
    acc = fmaf(kv.z, uv.z, acc);
    acc = fmaf(kv.w, uv.w, acc);
  }
  for (int off = 16; off; off >>= 1) acc += __shfl_xor(acc, off, 32);
  if (lane == 0) ws[WS_S + row] = acc + ws[WS_C0];
}

// ---------------------------------------------------------------------------
// k3: per-batch softmax over L=2048 in-place (2 elements/thread, 1024 threads)
// ---------------------------------------------------------------------------
__global__ void k3_softmax(float* __restrict__ ws) {
  int b = blockIdx.x, tx = threadIdx.x;
  float* s = ws + WS_S + b * L_;
  float v0 = s[tx], v1 = s[tx + 1024];
  __shared__ float red[32];
  int lane = tx & 31, wv = tx >> 5;

  float m = fmaxf(v0, v1);
  for (int off = 16; off; off >>= 1) m = fmaxf(m, __shfl_xor(m, off, 32));
  if (lane == 0) red[wv] = m;
  __syncthreads();
  if (wv == 0) {
    float t = red[lane];
    for (int off = 16; off; off >>= 1) t = fmaxf(t, __shfl_xor(t, off, 32));
    if (lane == 0) red[0] = t;
  }
  __syncthreads();
  m = red[0];
  __syncthreads();

  float e0 = expf(v0 - m), e1 = expf(v1 - m);
  float zs = e0 + e1;
  for (int off = 16; off; off >>= 1) zs += __shfl_xor(zs, off, 32);
  if (lane == 0) red[wv] = zs;
  __syncthreads();
  if (wv == 0) {
    float t = red[lane];
    for (int off = 16; off; off >>= 1) t += __shfl_xor(t, off, 32);
    if (lane == 0) red[0] = t;
  }
  __syncthreads();
  float inv = 1.0f / red[0];
  s[tx]        = e0 * inv;
  s[tx + 1024] = e1 * inv;
}

// ---------------------------------------------------------------------------
// k4: vbar[b,c] += sum over an L-chunk of p[b,l]*value[b,l,c]. float4 streams,
// atomic reduction across 32 L-chunks.  grid B*32 x 256
// ---------------------------------------------------------------------------
__global__ void k4_vbar(const float* __restrict__ value, float* __restrict__ ws) {
  int b = blockIdx.x >> 5;
  int chunk = blockIdx.x & 31;
  int tx = threadIdx.x;                            // covers c/4 = 0..255
  const float* p = ws + WS_S + b * L_;
  const float4* vb = (const float4*)(value + (size_t)b * L_ * CIN_);
  float ax = 0.f, ay = 0.f, az = 0.f, aw = 0.f;
  int l0 = chunk * 64;
  for (int l = l0; l < l0 + 64; ++l) {
    float pl = p[l];
    float4 v = vb[(size_t)l * (CIN_ / 4) + tx];
    ax = fmaf(pl, v.x, ax);
    ay = fmaf(pl, v.y, ay);
    az = fmaf(pl, v.z, az);
    aw = fmaf(pl, v.w, aw);
  }
  float* d = ws + WS_VBAR + b * CIN_ + tx * 4;
  atomicAdd(d + 0, ax);
  atomicAdd(d + 1, ay);
  atomicAdd(d + 2, az);
  atomicAdd(d + 3, aw);
}

// ---------------------------------------------------------------------------
// k5: out_row = vbar(16 x 1024, rows 8..15 = 0) @ Wv^T + bv
//     via V_WMMA_F32_16X16X4_F32.  One wave per 16-wide N tile; 64 waves.
// A frag (16x4 f32): lanes hold M=lane&15; VGPR p, half h -> K = k0 + p + 2h.
// B frag (4x16 f32): lanes hold N=lane&15; same K mapping.
// D (16x16 f32):     VGPR p, half h -> row M = p + 8h, col N = lane&15.
// Padded rows make A a direct load -> pure load+wmma inner loop (no masking).
// ---------------------------------------------------------------------------
__global__ void k5_wmma_out(const float* __restrict__ Wv,
                            const float* __restrict__ bv,
                            float* __restrict__ ws) {
  int tx = threadIdx.x;
  int lane = tx & 31, wv = tx >> 5;
  int gw = blockIdx.x * 8 + wv;                    // 0..63
  int n0 = gw * 16;
  int h = lane >> 4, li = lane & 15;

  const float* Arow = ws + WS_VBAR + (size_t)li * CIN_;   // rows 8..15 are zero
  const float* Brow = Wv + (size_t)(n0 + li) * CIN_;

  v8f acc = {};
  for (int k0 = 0; k0 < CIN_; k0 += 4) {
    int ka = k0 + 2 * h;
    v2f a, bm;
    a.x = Arow[ka];
    a.y = Arow[ka + 1];
    bm.x = Brow[ka];
    bm.y = Brow[ka + 1];
    acc = __builtin_amdgcn_wmma_f32_16x16x4_f32(
        /*neg_a=*/false, a, /*neg_b=*/false, bm,
        /*c_mod=*/(short)0, acc, /*reuse_a=*/false, /*reuse_b=*/false);
  }

  if (h == 0) {                                    // rows 0..7 = valid batches
    int n = n0 + li;
    float bias = bv[n];
    float* orow = ws + WS_OUTROW;
#pragma unroll
    for (int pi = 0; pi < 8; ++pi)
      orow[(size_t)pi * COUT_ + n] = acc[pi] + bias;
  }
}

// ---------------------------------------------------------------------------
// k6: broadcast out_row[b,:] to all q. One block per (b,q) row, float4 stores.
// grid B*L x 256
// ---------------------------------------------------------------------------
__global__ void k6_broadcast(const float* __restrict__ ws, float* __restrict__ out) {
  size_t row = blockIdx.x;                         // b*L + q
  int b = (int)(row >> 11);                        // L = 2048
  const float4* src = (const float4*)(ws + WS_OUTROW + (size_t)b * COUT_);
  float4* dst = (float4*)(out + row * (size_t)COUT_);
  dst[threadIdx.x] = src[threadIdx.x];
}

extern "C" void kernel_launch(void* const* d_in, const int* in_sizes, int n_in,
                              void* d_out, int out_size, void* d_ws, size_t ws_size,
                              hipStream_t stream) {
  (void)in_sizes; (void)n_in; (void)out_size; (void)ws_size;
  // setup_inputs order: query,key,value,Wq,bq,Wk,bk,Wv,bv,fcq_w,fcq_b,fck_w,fck_b
  const float* key   = (const float*)d_in[1];
  const float* value = (const float*)d_in[2];
  const float* Wk    = (const float*)d_in[5];
  const float* bk    = (const float*)d_in[6];
  const float* Wv    = (const float*)d_in[7];
  const float* bv    = (const float*)d_in[8];
  const float* fckw  = (const float*)d_in[11];
  const float* fckb  = (const float*)d_in[12];
  float* out = (float*)d_out;
  float* ws  = (float*)d_ws;

  k1_fold_key_weights<<<4, 256, 0, stream>>>(Wk, bk, fckw, fckb, ws);
  k2_scores<<<B_ * L_ / 8, 256, 0, stream>>>(key, ws);
  k3_softmax<<<B_, 1024, 0, stream>>>(ws);
  k4_vbar<<<B_ * 32, 256, 0, stream>>>(value, ws);
  k5_wmma_out<<<8, 256, 0, stream>>>(Wv, bv, ws);
  k6_broadcast<<<B_ * L_, 256, 0, stream>>>(ws, out);
}